// POIEncoder_47330539602639
// MI455X (gfx1250) — compile-verified
//
#include <hip/hip_runtime.h>

// GATConv(heads=1) + LayerNorm for MI455X (gfx1250), wave32 + WMMA fp32.

namespace {
constexpr int   N     = 100000;      // nodes
constexpr int   E     = 1600000;     // edges (before self loops)
constexpr int   ETOT  = E + N;       // edges + self loops
constexpr int   KDIM  = 128;         // IN_CH
constexpr int   H     = 64;          // HID
constexpr float SLOPE = 0.2f;
constexpr float EPS   = 1e-5f;
constexpr int   MTILES = N / 16;     // 6250 exact
constexpr int   WSTRIDE = KDIM + 4;  // 132: padded k-stride for transposed W in LDS
}

typedef float v2f __attribute__((ext_vector_type(2)));
typedef float v8f __attribute__((ext_vector_type(8)));

// ---------------------------------------------------------------------------
// K1: h = x @ W via V_WMMA_F32_16X16X4_F32.
// One wave per 16-row tile, 4 accumulators cover all 64 output columns.
// W staged TRANSPOSED in LDS (sWt[n*132 + k]) so each B fragment
// {W[k][n], W[k+1][n]} is a contiguous, 8B-aligned ds_load_b64.
// Padded stride 132 (== 4 mod 64 banks) -> conflict-free b64 reads.
// ---------------------------------------------------------------------------
__global__ __launch_bounds__(256) void gemm_xw_wmma(const float* __restrict__ x,
                                                    const float* __restrict__ W,
                                                    float* __restrict__ h) {
  __shared__ float sWt[H * WSTRIDE];
  // cooperative transpose: read W[k][n] coalesced (float4 over n), scatter to LDS
  for (int i = threadIdx.x; i < (KDIM * H) / 4; i += 256) {
    const int k  = i >> 4;          // 16 float4 per row of 64
    const int n4 = (i & 15) * 4;
    float4 w = ((const float4*)W)[i];
    sWt[(n4 + 0) * WSTRIDE + k] = w.x;
    sWt[(n4 + 1) * WSTRIDE + k] = w.y;
    sWt[(n4 + 2) * WSTRIDE + k] = w.z;
    sWt[(n4 + 3) * WSTRIDE + k] = w.w;
  }
  __syncthreads();

  const int wave = threadIdx.x >> 5;
  const int lane = threadIdx.x & 31;
  const int tile = blockIdx.x * 8 + wave;          // 16-row M tile
  if (tile >= MTILES) return;                      // wave-uniform: EXEC stays full

  const int g = lane >> 4;                         // lane group (0/1)
  const int m = lane & 15;                         // A: row in tile; B/C: column
  const float* xrow = x + (size_t)(tile * 16 + m) * KDIM;
  const float* wt0 = sWt + (size_t)(m + 0) * WSTRIDE + 2 * g;   // col m      (n-tile 0)
  const float* wt1 = sWt + (size_t)(m + 16) * WSTRIDE + 2 * g;  // col m+16   (n-tile 1)
  const float* wt2 = sWt + (size_t)(m + 32) * WSTRIDE + 2 * g;  // col m+32   (n-tile 2)
  const float* wt3 = sWt + (size_t)(m + 48) * WSTRIDE + 2 * g;  // col m+48   (n-tile 3)

  v8f acc0 = {}, acc1 = {}, acc2 = {}, acc3 = {};

  for (int k = 0; k < KDIM; k += 4) {
    // A 16x4 f32: lane l reg r holds A[m][k + 2*g + r] -> contiguous float2
    v2f a  = *(const v2f*)(xrow + k + 2 * g);
    // B 4x16 f32: lane l reg r holds B[k + 2*g + r][col] -> contiguous in sWt
    v2f b0 = *(const v2f*)(wt0 + k);
    v2f b1 = *(const v2f*)(wt1 + k);
    v2f b2 = *(const v2f*)(wt2 + k);
    v2f b3 = *(const v2f*)(wt3 + k);
    acc0 = __builtin_amdgcn_wmma_f32_16x16x4_f32(false, a, false, b0, (short)0, acc0, false, false);
    acc1 = __builtin_amdgcn_wmma_f32_16x16x4_f32(false, a, false, b1, (short)0, acc1, false, false);
    acc2 = __builtin_amdgcn_wmma_f32_16x16x4_f32(false, a, false, b2, (short)0, acc2, false, false);
    acc3 = __builtin_amdgcn_wmma_f32_16x16x4_f32(false, a, false, b3, (short)0, acc3, false, false);
  }

  // C/D layout: lane l, reg r -> row = r + 8*g, col = 16*t + m
  float* hb = h + (size_t)tile * 16 * H;
#pragma unroll
  for (int r = 0; r < 8; ++r) {
    float* o = hb + (size_t)(r + 8 * g) * H + m;
    o[0]  = acc0[r];
    o[16] = acc1[r];
    o[32] = acc2[r];
    o[48] = acc3[r];
  }
}

// ---------------------------------------------------------------------------
// K2: a_src[i] = h[i].att_src ; a_dst[i] = h[i].att_dst  (16 lanes per row)
// ---------------------------------------------------------------------------
__global__ __launch_bounds__(256) void attn_scores(const float* __restrict__ h,
                                                   const float* __restrict__ att_src,
                                                   const float* __restrict__ att_dst,
                                                   float* __restrict__ asrc,
                                                   float* __restrict__ adst) {
  const int t   = blockIdx.x * 256 + threadIdx.x;
  const int row = t >> 4;
  const int c   = t & 15;
  if (row >= N) return;
  float4 hv = *(const float4*)(h + (size_t)row * H + 4 * c);
  float4 as = *(const float4*)(att_src + 4 * c);
  float4 ad = *(const float4*)(att_dst + 4 * c);
  float ps = hv.x * as.x + hv.y * as.y + hv.z * as.z + hv.w * as.w;
  float pd = hv.x * ad.x + hv.y * ad.y + hv.z * ad.z + hv.w * ad.w;
#pragma unroll
  for (int off = 8; off >= 1; off >>= 1) {
    ps += __shfl_xor(ps, off);
    pd += __shfl_xor(pd, off);
  }
  if (c == 0) { asrc[row] = ps; adst[row] = pd; }
}

// Monotonic float<->uint mapping so atomicMax(u32) implements float max.
__device__ __forceinline__ unsigned fenc(float f) {
  unsigned u = __float_as_uint(f);
  return (u & 0x80000000u) ? ~u : (u | 0x80000000u);
}
__device__ __forceinline__ float fdec(unsigned k) {
  return __uint_as_float((k & 0x80000000u) ? (k ^ 0x80000000u) : ~k);
}

// ---------------------------------------------------------------------------
// K3: e = leaky_relu(a_src[s] + a_dst[d]); scatter-max over dst (L2 atomics)
// ---------------------------------------------------------------------------
__global__ __launch_bounds__(256) void edge_logits(const int* __restrict__ ei,
                                                   const float* __restrict__ asrc,
                                                   const float* __restrict__ adst,
                                                   float* __restrict__ ebuf,
                                                   unsigned* __restrict__ mkey) {
  const int i = blockIdx.x * 256 + threadIdx.x;
  if (i >= ETOT) return;
  int s, d;
  if (i < E) { s = ei[i]; d = ei[E + i]; } else { s = d = i - E; }
  float e = asrc[s] + adst[d];
  e = e > 0.0f ? e : SLOPE * e;
  ebuf[i] = e;
  atomicMax(mkey + d, fenc(e));
}

// ---------------------------------------------------------------------------
// K4: ex = exp(e - max[d]); scatter-sum denominators
// ---------------------------------------------------------------------------
__global__ __launch_bounds__(256) void edge_exp(const int* __restrict__ ei,
                                                const unsigned* __restrict__ mkey,
                                                float* __restrict__ ebuf,
                                                float* __restrict__ denom) {
  const int i = blockIdx.x * 256 + threadIdx.x;
  if (i >= ETOT) return;
  const int d = (i < E) ? ei[E + i] : (i - E);
  float ex = __expf(ebuf[i] - fdec(mkey[d]));
  ebuf[i] = ex;
  atomicAdd(denom + d, ex);
}

// ---------------------------------------------------------------------------
// K5: agg[d] += alpha * h[s]; 16 lanes per edge, float4 per lane, f32 atomics
// ---------------------------------------------------------------------------
__global__ __launch_bounds__(256) void aggregate(const int* __restrict__ ei,
                                                 const float* __restrict__ ebuf,
                                                 const float* __restrict__ denom,
                                                 const float* __restrict__ h,
                                                 float* __restrict__ agg) {
  const int t   = blockIdx.x * 256 + threadIdx.x;
  const int eid = t >> 4;
  const int c   = t & 15;
  if (eid >= ETOT) return;
  int s, d;
  if (eid < E) { s = ei[eid]; d = ei[E + eid]; } else { s = d = eid - E; }
  const float alpha = ebuf[eid] / denom[d];
  float4 hv = *(const float4*)(h + (size_t)s * H + 4 * c);
  float* o = agg + (size_t)d * H + 4 * c;
  atomicAdd(o + 0, hv.x * alpha);
  atomicAdd(o + 1, hv.y * alpha);
  atomicAdd(o + 2, hv.z * alpha);
  atomicAdd(o + 3, hv.w * alpha);
}

// ---------------------------------------------------------------------------
// K6: out = LayerNorm(agg + bias) * gamma + beta  (16 lanes per row)
// ---------------------------------------------------------------------------
__global__ __launch_bounds__(256) void layernorm(const float* __restrict__ agg,
                                                 const float* __restrict__ bias,
                                                 const float* __restrict__ gamma,
                                                 const float* __restrict__ beta,
                                                 float* __restrict__ out) {
  const int t   = blockIdx.x * 256 + threadIdx.x;
  const int row = t >> 4;
  const int c   = t & 15;
  if (row >= N) return;
  float4 v = *(const float4*)(agg + (size_t)row * H + 4 * c);
  float4 b = *(const float4*)(bias + 4 * c);
  v.x += b.x; v.y += b.y; v.z += b.z; v.w += b.w;

  float s = v.x + v.y + v.z + v.w;
#pragma unroll
  for (int off = 8; off >= 1; off >>= 1) s += __shfl_xor(s, off);
  const float mu = s * (1.0f / H);

  float dx = v.x - mu, dy = v.y - mu, dz = v.z - mu, dw = v.w - mu;
  float sq = dx * dx + dy * dy + dz * dz + dw * dw;
#pragma unroll
  for (int off = 8; off >= 1; off >>= 1) sq += __shfl_xor(sq, off);
  const float rstd = rsqrtf(sq * (1.0f / H) + EPS);

  float4 gm = *(const float4*)(gamma + 4 * c);
  float4 bt = *(const float4*)(beta + 4 * c);
  float4 r;
  r.x = dx * rstd * gm.x + bt.x;
  r.y = dy * rstd * gm.y + bt.y;
  r.z = dz * rstd * gm.z + bt.z;
  r.w = dw * rstd * gm.w + bt.w;
  *(float4*)(out + (size_t)row * H + 4 * c) = r;
}

// ---------------------------------------------------------------------------
extern "C" void kernel_launch(void* const* d_in, const int* in_sizes, int n_in,
                              void* d_out, int out_size, void* d_ws, size_t ws_size,
                              hipStream_t stream) {
  const float* x     = (const float*)d_in[0];
  const int*   ei    = (const int*)d_in[1];   // int32 (JAX x64 off)
  const float* W     = (const float*)d_in[2];
  const float* att_s = (const float*)d_in[3];
  const float* att_d = (const float*)d_in[4];
  const float* bias  = (const float*)d_in[5];
  const float* gamma = (const float*)d_in[6];
  const float* beta  = (const float*)d_in[7];
  float* out = (float*)d_out;

  // workspace layout (all 4-byte elements; bases stay 16B aligned)
  float*    h     = (float*)d_ws;                   // N*H
  float*    agg   = h + (size_t)N * H;              // N*H  (accumulated -> zero)
  float*    asrc  = agg + (size_t)N * H;            // N
  float*    adst  = asrc + N;                       // N
  unsigned* mkey  = (unsigned*)(adst + N);          // N    (accumulated -> zero)
  float*    denom = (float*)(mkey + N);             // N    (accumulated -> zero)
  float*    ebuf  = denom + N;                      // ETOT

  // zero the accumulation buffers every call (mkey==0 acts as -inf key)
  hipMemsetAsync(agg, 0, (size_t)N * H * sizeof(float), stream);
  hipMemsetAsync(mkey, 0, 2 * (size_t)N * sizeof(unsigned), stream);

  gemm_xw_wmma<<<(MTILES + 7) / 8, 256, 0, stream>>>(x, W, h);
  attn_scores<<<(N * 16 + 255) / 256, 256, 0, stream>>>(h, att_s, att_d, asrc, adst);
  edge_logits<<<(ETOT + 255) / 256, 256, 0, stream>>>(ei, asrc, adst, ebuf, mkey);
  edge_exp<<<(ETOT + 255) / 256, 256, 0, stream>>>(ei, mkey, ebuf, denom);
  aggregate<<<(ETOT * 16 + 255) / 256, 256, 0, stream>>>(ei, ebuf, denom, h, agg);
  layernorm<<<(N * 16 + 255) / 256, 256, 0, stream>>>(agg, bias, gamma, beta, out);
}